// FeatureSimilarity_32658931319006
// MI455X (gfx1250) — compile-verified
//
#include <hip/hip_runtime.h>

typedef float v2f __attribute__((ext_vector_type(2)));
typedef float v4f __attribute__((ext_vector_type(4)));
typedef float v8f __attribute__((ext_vector_type(8)));

#define TILE        128
#define D_DIM       128
#define LDS_STRIDE  132   // 128 + 4 pad: bank = (4*M + k) & 63 -> conflict-free b64 loads

__global__ __launch_bounds__(256, 2)
void feat_sim_l2_kernel(const float* __restrict__ F,
                        float* __restrict__ out,
                        int N) {
    __shared__ float ldsA[TILE * LDS_STRIDE];
    __shared__ float ldsB[TILE * LDS_STRIDE];
    __shared__ float sqA[TILE];
    __shared__ float sqB[TILE];

    const int tid   = threadIdx.x;
    const int iBase = blockIdx.y * TILE;
    const int jBase = blockIdx.x * TILE;

    // ---- stage A (rows iBase..) and B (rows jBase..) tiles into LDS, float4 ----
    for (int idx = tid; idx < TILE * (D_DIM / 4); idx += 256) {
        const int row = idx >> 5;     // / (128/4)
        const int c4  = idx & 31;
        v4f a = *(const v4f*)(F + (size_t)(iBase + row) * D_DIM + c4 * 4);
        v4f b = *(const v4f*)(F + (size_t)(jBase + row) * D_DIM + c4 * 4);
        *(v4f*)(ldsA + row * LDS_STRIDE + c4 * 4) = a;
        *(v4f*)(ldsB + row * LDS_STRIDE + c4 * 4) = b;
    }
    __syncthreads();

    // ---- row squared norms (threads 0-127 -> A, 128-255 -> B) ----
    {
        const int r = tid & 127;
        const float* p = (tid < 128) ? (ldsA + r * LDS_STRIDE)
                                     : (ldsB + r * LDS_STRIDE);
        float s = 0.0f;
        #pragma unroll
        for (int k = 0; k < D_DIM; ++k) s = fmaf(p[k], p[k], s);
        if (tid < 128) sqA[r] = s; else sqB[r] = s;
    }
    __syncthreads();

    // ---- wave tiling: 8 waves, each owns 4 M-subtiles x 2 N-subtiles of 16x16 ----
    const int lane  = tid & 31;
    const int warp  = tid >> 5;
    const int waveM = warp & 1;      // 0..1 -> M-subtiles [4*waveM, 4*waveM+3]
    const int waveN = warp >> 1;     // 0..3 -> N-subtiles [2*waveN, 2*waveN+1]

    // ISA 7.12.2: 32-bit A 16x4 -> lane holds M = lane&15, K = k0 + 2*(lane>>4) + {0,1}
    const int mRow  = lane & 15;
    const int kHalf = (lane >> 4) << 1;

    v8f acc[4][2];
    #pragma unroll
    for (int mi = 0; mi < 4; ++mi)
        #pragma unroll
        for (int ni = 0; ni < 2; ++ni)
            acc[mi][ni] = (v8f)0.0f;

    #pragma unroll 2
    for (int k = 0; k < D_DIM; k += 4) {
        v2f afrag[4];
        v2f bfrag[2];
        #pragma unroll
        for (int mi = 0; mi < 4; ++mi) {
            const int row = (waveM * 4 + mi) * 16 + mRow;
            afrag[mi] = *(const v2f*)(ldsA + row * LDS_STRIDE + k + kHalf);
        }
        #pragma unroll
        for (int ni = 0; ni < 2; ++ni) {
            const int row = (waveN * 2 + ni) * 16 + mRow;   // B = F^T: same layout
            bfrag[ni] = *(const v2f*)(ldsB + row * LDS_STRIDE + k + kHalf);
        }
        #pragma unroll
        for (int mi = 0; mi < 4; ++mi)
            #pragma unroll
            for (int ni = 0; ni < 2; ++ni)
                acc[mi][ni] = __builtin_amdgcn_wmma_f32_16x16x4_f32(
                    /*neg_a=*/false, afrag[mi],
                    /*neg_b=*/false, bfrag[ni],
                    /*c_mod=*/(short)0, acc[mi][ni],
                    /*reuse_a=*/false, /*reuse_b=*/false);
    }

    // ---- epilogue: -sqrt(max(0, ||a||^2 + ||b||^2 - 2*dot)) ----
    // C/D layout: VGPR v -> M = v + 8*(lane>>4), N = lane&15
    const int    rowHalf = (lane >> 4) << 3;
    const size_t stride  = (size_t)N;
    #pragma unroll
    for (int mi = 0; mi < 4; ++mi) {
        const int subM = (waveM * 4 + mi) * 16;
        #pragma unroll
        for (int ni = 0; ni < 2; ++ni) {
            const int subN = (waveN * 2 + ni) * 16;
            const int col  = subN + (lane & 15);
            const float sj = sqB[col];
            // walk the column pointer: avoids a 64-bit mul per store
            float* p = out + (size_t)(iBase + subM + rowHalf) * stride
                           + (size_t)(jBase + col);
            #pragma unroll
            for (int v = 0; v < 8; ++v) {
                const int m  = subM + v + rowHalf;
                float d2 = sqA[m] + sj - 2.0f * acc[mi][ni][v];
                d2 = fmaxf(d2, 0.0f);
                *p = -__builtin_amdgcn_sqrtf(d2);   // raw v_sqrt_f32 (~1 ulp, inputs clamped >= 0)
                p += stride;
            }
        }
    }
}

extern "C" void kernel_launch(void* const* d_in, const int* in_sizes, int n_in,
                              void* d_out, int out_size, void* d_ws, size_t ws_size,
                              hipStream_t stream) {
    const float* F = (const float*)d_in[0];
    float* out     = (float*)d_out;
    const int N = in_sizes[0] / D_DIM;          // 8192

    dim3 grid(N / TILE, N / TILE);              // 64 x 64
    dim3 block(256);                            // 8 waves (wave32)
    feat_sim_l2_kernel<<<grid, block, 0, stream>>>(F, out, N);
}